// HGCNDecoder_16415365005392
// MI455X (gfx1250) — compile-verified
//
#include <hip/hip_runtime.h>
#include <hip/hip_bf16.h>

// ---------------------------------------------------------------------------
// HGCN decoder for MI455X (gfx1250), fp32 end-to-end.
// GEMMs use V_WMMA_F32_16X16X4_F32 (fp32 matrix pipe, wave32).
// Edge aggregation uses global_atomic_add_f32 (agg/xt are L2-resident: 8MB).
// ---------------------------------------------------------------------------

typedef __attribute__((ext_vector_type(2))) float v2f;
typedef __attribute__((ext_vector_type(8))) float v8f;

#define EPS_F      1e-15f
#define MAX_TANH_F 15.0f
#define ARTANH_LIM (1.0f - 1e-5f)
#define MAXN_F     (1.0f - 4e-3f)   // (1 - PROJ_EPS)/sqrt(c), c == 1
#define DIM        64

__device__ __forceinline__ float tanh_c(float x) {
    x = fminf(fmaxf(x, -MAX_TANH_F), MAX_TANH_F);
    return tanhf(x);
}

__device__ __forceinline__ float artanh_c(float x) {
    x = fminf(fmaxf(x, -ARTANH_LIM), ARTANH_LIM);
    return 0.5f * logf((1.0f + x) / (1.0f - x));
}

__device__ __forceinline__ float wave_sum(float v) {
    // wave32 butterfly reduction
    #pragma unroll
    for (int m = 16; m >= 1; m >>= 1) v += __shfl_xor(v, m, 32);
    return v;
}

__device__ __forceinline__ float cnorm(float sumsq) {
    return fmaxf(sqrtf(sumsq), EPS_F);
}

// proj() onto Poincare ball; lane owns (a, b). n is wave-uniform -> no
// divergence around the branch.
__device__ __forceinline__ void proj_pair(float& a, float& b) {
    float n = cnorm(wave_sum(a * a + b * b));
    if (n > MAXN_F) { float s = MAXN_F / n; a *= s; b *= s; }
}

// ---------------------------------------------------------------------------
// Kernel: x = proj(expmap0(h))   (one wave per row)
// ---------------------------------------------------------------------------
__global__ void init_expmap_kernel(const float* __restrict__ H,
                                   float* __restrict__ X, int n_nodes) {
    int row  = blockIdx.x * (blockDim.x >> 5) + (threadIdx.x >> 5);
    if (row >= n_nodes) return;
    int lane = threadIdx.x & 31;
    const float* hr = H + (size_t)row * DIM;
    float u0 = hr[lane], u1 = hr[lane + 32];
    float n = cnorm(wave_sum(u0 * u0 + u1 * u1));
    float f = tanh_c(n) / n;
    u0 *= f; u1 *= f;
    proj_pair(u0, u1);
    X[(size_t)row * DIM + lane]      = u0;
    X[(size_t)row * DIM + lane + 32] = u1;
}

// ---------------------------------------------------------------------------
// Kernel: hb = proj(expmap0(b))  (single wave)
// ---------------------------------------------------------------------------
__global__ void prep_hb_kernel(const float* __restrict__ b,
                               float* __restrict__ hb) {
    int lane = threadIdx.x;
    float u0 = b[lane], u1 = b[lane + 32];
    float n = cnorm(wave_sum(u0 * u0 + u1 * u1));
    float f = tanh_c(n) / n;
    u0 *= f; u1 *= f;
    proj_pair(u0, u1);
    hb[lane] = u0; hb[lane + 32] = u1;
}

// ---------------------------------------------------------------------------
// Kernel: MX[N,64] = X[N,64] @ W^T, W row-major [64,64]  (fp32 WMMA)
// One wave per 16x16 output tile, K=64 in 16 chained 16x16x4 WMMAs.
// Fragment layouts per CDNA5 ISA 7.12.2:
//   A 16x4 f32 : VGPR0 = {K0 | K2}, VGPR1 = {K1 | K3}, lanes 0-15 = M,
//                lanes 16-31 = M with K+2.
//   B 4x16 f32 : rows striped across lanes (mirrored).
//   C/D        : VGPR r = rows r (lanes 0-15) and r+8 (lanes 16-31).
// ---------------------------------------------------------------------------
__global__ void gemm64_wmma_kernel(const float* __restrict__ X,
                                   const float* __restrict__ W,
                                   float* __restrict__ MX, int n_nodes) {
    int wave = blockIdx.x * (blockDim.x >> 5) + (threadIdx.x >> 5);
    int tile_row = wave >> 2;       // N/16 row tiles
    int tile_col = wave & 3;        // 64/16 = 4 col tiles
    if (tile_row * 16 >= n_nodes) return;   // wave-uniform
    int lane = threadIdx.x & 31;
    int half = lane >> 4;           // 0 or 1
    int lid  = lane & 15;
    int row0 = tile_row * 16, col0 = tile_col * 16;

    const float* xrow = X + (size_t)(row0 + lid) * DIM;  // A row (M = lid)
    const float* wrow = W + (size_t)(col0 + lid) * DIM;  // B col (N = lid), B[k][n] = W[n][k]

    v8f acc = {};
    #pragma unroll
    for (int k0 = 0; k0 < DIM; k0 += 4) {
        v2f a, bf;
        a.x  = xrow[k0 + 2 * half];
        a.y  = xrow[k0 + 2 * half + 1];
        bf.x = wrow[k0 + 2 * half];
        bf.y = wrow[k0 + 2 * half + 1];
        acc = __builtin_amdgcn_wmma_f32_16x16x4_f32(
            false, a, false, bf, (short)0, acc, false, false);
    }
    #pragma unroll
    for (int r = 0; r < 8; ++r) {
        int row = row0 + r + 8 * half;
        MX[(size_t)row * DIM + col0 + lid] = acc[r];
    }
}

// ---------------------------------------------------------------------------
// Kernel: per-row HypLinear tail:
//   mv  = proj(mobius_matvec post-scale)
//   res = proj(mobius_add(mv, hb))
//   xt  = logmap0(res)
// ---------------------------------------------------------------------------
__global__ void post_mv_kernel(const float* __restrict__ X,
                               const float* __restrict__ MX,
                               const float* __restrict__ hb,
                               float* __restrict__ XT, int n_nodes) {
    int row = blockIdx.x * (blockDim.x >> 5) + (threadIdx.x >> 5);
    if (row >= n_nodes) return;
    int lane = threadIdx.x & 31;
    const float* xr = X  + (size_t)row * DIM;
    const float* mr = MX + (size_t)row * DIM;
    float x0 = xr[lane], x1 = xr[lane + 32];
    float m0 = mr[lane], m1 = mr[lane + 32];

    // mobius_matvec: tanh(mxn/xn * artanh(xn)) * mx / mxn   (sc = 1)
    float xn  = cnorm(wave_sum(x0 * x0 + x1 * x1));
    float mxn = cnorm(wave_sum(m0 * m0 + m1 * m1));
    float f = tanh_c(mxn / xn * artanh_c(xn)) / mxn;
    m0 *= f; m1 *= f;
    proj_pair(m0, m1);

    // mobius_add(mv, hb)
    float y0 = hb[lane], y1 = hb[lane + 32];
    float x2 = wave_sum(m0 * m0 + m1 * m1);
    float y2 = wave_sum(y0 * y0 + y1 * y1);
    float xy = wave_sum(m0 * y0 + m1 * y1);
    float ca  = 1.0f + 2.0f * xy + y2;
    float cb  = 1.0f - x2;
    float den = fmaxf(1.0f + 2.0f * xy + x2 * y2, EPS_F);
    float r0 = (ca * m0 + cb * y0) / den;
    float r1 = (ca * m1 + cb * y1) / den;
    proj_pair(r0, r1);

    // logmap0
    float rn = cnorm(wave_sum(r0 * r0 + r1 * r1));
    float g = artanh_c(rn) / rn;
    XT[(size_t)row * DIM + lane]      = g * r0;
    XT[(size_t)row * DIM + lane + 32] = g * r1;
}

// ---------------------------------------------------------------------------
// Kernel: zero fill (grid-stride)
// ---------------------------------------------------------------------------
__global__ void zero_kernel(float* __restrict__ p, size_t n) {
    size_t i = (size_t)blockIdx.x * blockDim.x + threadIdx.x;
    size_t stride = (size_t)gridDim.x * blockDim.x;
    for (; i < n; i += stride) p[i] = 0.0f;
}

// ---------------------------------------------------------------------------
// Kernel: edge scatter — agg[rows[e]] += xt[cols[e]] * edge_mask[e]
// One wave per edge (lane l owns dims l, l+32). Atomics hit L2 (agg is 8MB).
// ---------------------------------------------------------------------------
__global__ void scatter_kernel(const float* __restrict__ XT,
                               const int* __restrict__ rows,
                               const int* __restrict__ cols,
                               const float* __restrict__ edge_mask,
                               float* __restrict__ AGG, int n_edges) {
    int e = blockIdx.x * (blockDim.x >> 5) + (threadIdx.x >> 5);
    if (e >= n_edges) return;   // wave-uniform
    int lane = threadIdx.x & 31;
    int r = rows[e], c = cols[e];
    float m = edge_mask[e];
    const float* src = XT  + (size_t)c * DIM;
    float*       dst = AGG + (size_t)r * DIM;
    atomicAdd(&dst[lane],      src[lane]      * m);
    atomicAdd(&dst[lane + 32], src[lane + 32] * m);
}

// ---------------------------------------------------------------------------
// Kernel: per-row HypAgg/HypAct tail:
//   a = agg*node_mask; x = proj(expmap0(a));
//   x = proj(expmap0(relu(logmap0(x)), c_out))
// ---------------------------------------------------------------------------
__global__ void post_agg_kernel(const float* __restrict__ AGG,
                                const float* __restrict__ node_mask,
                                float* __restrict__ X, int n_nodes) {
    int row = blockIdx.x * (blockDim.x >> 5) + (threadIdx.x >> 5);
    if (row >= n_nodes) return;
    int lane = threadIdx.x & 31;
    float nm = node_mask[row];
    float a0 = AGG[(size_t)row * DIM + lane]      * nm;
    float a1 = AGG[(size_t)row * DIM + lane + 32] * nm;

    // expmap0 + proj
    float n = cnorm(wave_sum(a0 * a0 + a1 * a1));
    float f = tanh_c(n) / n;
    a0 *= f; a1 *= f;
    proj_pair(a0, a1);

    // relu(logmap0(x))
    float ln = cnorm(wave_sum(a0 * a0 + a1 * a1));
    float g = artanh_c(ln) / ln;
    float t0 = fmaxf(g * a0, 0.0f);
    float t1 = fmaxf(g * a1, 0.0f);

    // expmap0 (c_out) + proj
    float tn = cnorm(wave_sum(t0 * t0 + t1 * t1));
    float f2 = tanh_c(tn) / tn;
    t0 *= f2; t1 *= f2;
    proj_pair(t0, t1);

    X[(size_t)row * DIM + lane]      = t0;
    X[(size_t)row * DIM + lane + 32] = t1;
}

// ---------------------------------------------------------------------------
// Kernel: out[N,16] = X[N,64] @ Wout^T + bout   (fp32 WMMA, Wout is [16,64])
// One wave per 16-row tile (all 16 output cols in one tile).
// ---------------------------------------------------------------------------
__global__ void gemm_out_wmma_kernel(const float* __restrict__ X,
                                     const float* __restrict__ Wout,
                                     const float* __restrict__ bout,
                                     float* __restrict__ OUT, int n_nodes) {
    int wave = blockIdx.x * (blockDim.x >> 5) + (threadIdx.x >> 5);
    if (wave * 16 >= n_nodes) return;   // wave-uniform
    int lane = threadIdx.x & 31;
    int half = lane >> 4;
    int lid  = lane & 15;
    int row0 = wave * 16;

    const float* xrow = X    + (size_t)(row0 + lid) * DIM;
    const float* wrow = Wout + (size_t)lid * DIM;   // B[k][n] = Wout[n][k]

    v8f acc = {};
    #pragma unroll
    for (int k0 = 0; k0 < DIM; k0 += 4) {
        v2f a, bf;
        a.x  = xrow[k0 + 2 * half];
        a.y  = xrow[k0 + 2 * half + 1];
        bf.x = wrow[k0 + 2 * half];
        bf.y = wrow[k0 + 2 * half + 1];
        acc = __builtin_amdgcn_wmma_f32_16x16x4_f32(
            false, a, false, bf, (short)0, acc, false, false);
    }
    float bias = bout[lid];
    #pragma unroll
    for (int r = 0; r < 8; ++r) {
        int row = row0 + r + 8 * half;
        OUT[(size_t)row * 16 + lid] = acc[r] + bias;
    }
}

// ---------------------------------------------------------------------------
// Host launcher
// ---------------------------------------------------------------------------
extern "C" void kernel_launch(void* const* d_in, const int* in_sizes, int n_in,
                              void* d_out, int out_size, void* d_ws, size_t ws_size,
                              hipStream_t stream) {
    const float* h         = (const float*)d_in[0];
    // d_in[1] = distances (unused by the reference)
    const int*   rows      = (const int*)d_in[2];
    const int*   cols      = (const int*)d_in[3];
    const float* node_mask = (const float*)d_in[4];
    const float* edge_mask = (const float*)d_in[5];
    const float* W0        = (const float*)d_in[6];
    const float* b0        = (const float*)d_in[7];
    const float* W1        = (const float*)d_in[8];
    const float* b1        = (const float*)d_in[9];
    const float* Wout      = (const float*)d_in[10];
    const float* bout      = (const float*)d_in[11];
    float*       out       = (float*)d_out;

    const int N  = in_sizes[0] / DIM;   // 32768
    const int nE = in_sizes[2];         // 1015808

    float* xbuf   = (float*)d_ws;
    float* mxbuf  = xbuf  + (size_t)N * DIM;
    float* xtbuf  = mxbuf + (size_t)N * DIM;
    float* aggbuf = xtbuf + (size_t)N * DIM;
    float* hbbuf  = aggbuf + (size_t)N * DIM;

    const int WPB = 8;                  // waves per 256-thread block
    dim3 blk(256);
    dim3 row_grid((N + WPB - 1) / WPB);                 // wave-per-row kernels
    dim3 gemm_grid((N / 16 * 4 + WPB - 1) / WPB);       // 16x16 tiles, 4 col tiles
    dim3 gemmo_grid((N / 16 + WPB - 1) / WPB);          // 16-row tiles
    dim3 edge_grid((nE + WPB - 1) / WPB);
    dim3 zero_grid(2048);

    // x = proj(expmap0(h))
    init_expmap_kernel<<<row_grid, blk, 0, stream>>>(h, xbuf, N);

    const float* Ws[2] = {W0, W1};
    const float* bs[2] = {b0, b1};
    for (int layer = 0; layer < 2; ++layer) {
        // mx = x @ W^T     (WMMA fp32)
        gemm64_wmma_kernel<<<gemm_grid, blk, 0, stream>>>(xbuf, Ws[layer], mxbuf, N);
        // hb = proj(expmap0(b))
        prep_hb_kernel<<<dim3(1), dim3(32), 0, stream>>>(bs[layer], hbbuf);
        // xt = logmap0(proj(mobius_add(proj(matvec), hb)))
        post_mv_kernel<<<row_grid, blk, 0, stream>>>(xbuf, mxbuf, hbbuf, xtbuf, N);
        // agg = scatter-add over edges
        zero_kernel<<<zero_grid, blk, 0, stream>>>(aggbuf, (size_t)N * DIM);
        scatter_kernel<<<edge_grid, blk, 0, stream>>>(xtbuf, rows, cols, edge_mask,
                                                      aggbuf, nE);
        // x = proj(expmap0(relu(logmap0(proj(expmap0(agg*mask))))))
        post_agg_kernel<<<row_grid, blk, 0, stream>>>(aggbuf, node_mask, xbuf, N);
    }

    // out = x @ Wout^T + bout    (WMMA fp32)
    gemm_out_wmma_kernel<<<gemmo_grid, blk, 0, stream>>>(xbuf, Wout, bout, out, N);
}